// KANLayer_69286412419131
// MI455X (gfx1250) — compile-verified
//
#include <hip/hip_runtime.h>

typedef float v2f __attribute__((ext_vector_type(2)));
typedef float v8f __attribute__((ext_vector_type(8)));

#define BSZ 64
#define ISZ 256
#define OSZ 256
#define BIG (BSZ * OSZ * ISZ)   // 4194304
#define OUT0 0
#define PRE0 (BSZ * OSZ)        // 16384
#define PACT0 (PRE0 + BIG)
#define PSPL0 (PRE0 + 2 * BIG)

// Zero the reduction output region (atomics accumulate into it each call).
__global__ __launch_bounds__(256) void kan_zero_out(float* __restrict__ out) {
    out[blockIdx.x * 256 + threadIdx.x] = 0.0f;
}

__global__ __launch_bounds__(256) void kan_main(
    const float* __restrict__ x,     // (B, I)
    const float* __restrict__ grid,  // (S, 6)  all rows identical
    const float* __restrict__ coef,  // (S, 8)
    const float* __restrict__ rs,    // (S,)
    const float* __restrict__ ss,    // (S,)
    const float* __restrict__ mk,    // (S,)
    float* __restrict__ dout) {
    // spline tile [b_local][o_local][i_local], i-stride 33 to dodge bank conflicts
    __shared__ float spl[16][16][33];
    __shared__ float silu_t[16][33];

    const int bid  = blockIdx.x;
    const int iseg = bid & 7;            // 8 segments of 32 i
    const int ot   = (bid >> 3) & 15;    // 16 o tiles
    const int bt   = bid >> 7;           // 4 b tiles
    const int i0     = iseg * 32;
    const int o_base = ot * 16;
    const int b_base = bt * 16;

    const int tid  = threadIdx.x;
    const int wave = tid >> 5;
    const int lane = tid & 31;
    const int half = lane >> 4;          // 0: lanes 0-15, 1: lanes 16-31
    const int bl   = lane & 15;

    // ---- Knot vector: grid row 0 extended by k=3 on each side ----
    const float g0 = grid[0];
    const float h  = (grid[5] - g0) * 0.2f;
    float t[12];
#pragma unroll
    for (int j = 0; j < 12; ++j) {
        t[j] = (j >= 3 && j < 9) ? grid[j - 3] : (g0 + (float)(j - 3) * h);
    }

    // i-invariant reciprocal knot gaps: invD[kk-1][j] = 1 / (t[j+kk] - t[j]).
    // Hoisted so the per-i Cox-de Boor recursion is pure FMA (no divisions).
    float invD1[11], invD2[10], invD3[9];
#pragma unroll
    for (int j = 0; j < 11; ++j) invD1[j] = 1.0f / (t[j + 1] - t[j]);
#pragma unroll
    for (int j = 0; j < 10; ++j) invD2[j] = 1.0f / (t[j + 2] - t[j]);
#pragma unroll
    for (int j = 0; j < 9;  ++j) invD3[j] = 1.0f / (t[j + 3] - t[j]);

    // ================= Phase 1: WMMA spline tiles =================
#pragma unroll
    for (int jj = 0; jj < 4; ++jj) {
        const int il = wave * 4 + jj;    // 0..31 across the 8 waves
        const int ig = i0 + il;

        const float xv = x[(b_base + bl) * ISZ + ig];
        const float sv = xv / (1.0f + __expf(-xv));   // silu
        silu_t[bl][il] = sv;             // both halves write identical value

        // Cox-de Boor, order 3: 11 -> 10 -> 9 -> 8 basis functions
        float Bv[11];
#pragma unroll
        for (int j = 0; j < 11; ++j)
            Bv[j] = (xv >= t[j] && xv < t[j + 1]) ? 1.0f : 0.0f;
        // kk = 1
#pragma unroll
        for (int j = 0; j < 10; ++j) {
            const float left  = (xv - t[j])     * invD1[j];
            const float right = (t[j + 2] - xv) * invD1[j + 1];
            Bv[j] = left * Bv[j] + right * Bv[j + 1];
        }
        // kk = 2
#pragma unroll
        for (int j = 0; j < 9; ++j) {
            const float left  = (xv - t[j])     * invD2[j];
            const float right = (t[j + 3] - xv) * invD2[j + 1];
            Bv[j] = left * Bv[j] + right * Bv[j + 1];
        }
        // kk = 3
#pragma unroll
        for (int j = 0; j < 8; ++j) {
            const float left  = (xv - t[j])     * invD3[j];
            const float right = (t[j + 4] - xv) * invD3[j + 1];
            Bv[j] = left * Bv[j] + right * Bv[j + 1];
        }

        // B-matrix (4x16, K x N=b): VGPR0 = K 0|2 per half, VGPR1 = K 1|3
        v2f bf0, bf1;
        bf0.x = Bv[2 * half];     bf0.y = Bv[2 * half + 1];
        bf1.x = Bv[4 + 2 * half]; bf1.y = Bv[4 + 2 * half + 1];

        // A-matrix (16x4, M=o x K): lane holds row o_base+bl; VGPR0 = K 0|2, VGPR1 = K 1|3
        const int orow = o_base + bl;
        const float* crow = coef + ((size_t)(orow * ISZ + ig)) * 8;
        const float2 cA = *(const float2*)(crow + 2 * half);
        const float2 cB = *(const float2*)(crow + 4 + 2 * half);
        v2f af0, af1;
        af0.x = cA.x; af0.y = cA.y;
        af1.x = cB.x; af1.y = cB.y;

        v8f acc = {0.f, 0.f, 0.f, 0.f, 0.f, 0.f, 0.f, 0.f};
        acc = __builtin_amdgcn_wmma_f32_16x16x4_f32(false, af0, false, bf0,
                                                    (short)0, acc, false, false);
        acc = __builtin_amdgcn_wmma_f32_16x16x4_f32(false, af1, false, bf1,
                                                    (short)0, acc, false, false);

        // D element (M=r+8*half, N=bl) -> spl[b][o][i]
#pragma unroll
        for (int r = 0; r < 8; ++r)
            spl[bl][r + 8 * half][il] = acc[r];
    }
    __syncthreads();

    // ============ Phase 2: coalesced stores (1 cacheline/lane) ============
    const int o_l = tid & 15;
    const int b_l = tid >> 4;
    const int og = o_base + o_l;
    const int bg = b_base + b_l;
    const size_t row = ((size_t)bg * OSZ + og) * (size_t)ISZ + (size_t)i0;

    float* __restrict__ pre  = dout + PRE0;
    float* __restrict__ pact = dout + PACT0;
    float* __restrict__ pspl = dout + PSPL0;

    const float4* __restrict__ x4  = (const float4*)(x  + bg * ISZ + i0);
    const float4* __restrict__ rs4 = (const float4*)(rs + og * ISZ + i0);
    const float4* __restrict__ ss4 = (const float4*)(ss + og * ISZ + i0);
    const float4* __restrict__ mk4 = (const float4*)(mk + og * ISZ + i0);

    float accsum = 0.0f;
#pragma unroll
    for (int c = 0; c < 8; ++c) {
        const float4 xv = x4[c];
        const float4 r4 = rs4[c];
        const float4 s4 = ss4[c];
        const float4 m4 = mk4[c];

        float4 sp, pa;
        sp.x = spl[b_l][o_l][4 * c + 0];
        sp.y = spl[b_l][o_l][4 * c + 1];
        sp.z = spl[b_l][o_l][4 * c + 2];
        sp.w = spl[b_l][o_l][4 * c + 3];
        const float si0 = silu_t[b_l][4 * c + 0];
        const float si1 = silu_t[b_l][4 * c + 1];
        const float si2 = silu_t[b_l][4 * c + 2];
        const float si3 = silu_t[b_l][4 * c + 3];

        pa.x = m4.x * (r4.x * si0 + s4.x * sp.x);
        pa.y = m4.y * (r4.y * si1 + s4.y * sp.y);
        pa.z = m4.z * (r4.z * si2 + s4.z * sp.z);
        pa.w = m4.w * (r4.w * si3 + s4.w * sp.w);

        *(float4*)(pre  + row + 4 * c) = xv;   // pre_activation = broadcast x
        *(float4*)(pact + row + 4 * c) = pa;   // post_activation
        *(float4*)(pspl + row + 4 * c) = sp;   // post_spline

        accsum += pa.x + pa.y + pa.z + pa.w;
    }

    // ============ Phase 3: i-reduction into out[b,o] ============
    atomicAdd(&dout[OUT0 + bg * OSZ + og], accsum);
}

extern "C" void kernel_launch(void* const* d_in, const int* in_sizes, int n_in,
                              void* d_out, int out_size, void* d_ws, size_t ws_size,
                              hipStream_t stream) {
    const float* x    = (const float*)d_in[0];
    const float* grid = (const float*)d_in[1];
    const float* coef = (const float*)d_in[2];
    const float* rs   = (const float*)d_in[3];
    const float* ss   = (const float*)d_in[4];
    const float* mk   = (const float*)d_in[5];
    float* out = (float*)d_out;

    kan_zero_out<<<dim3(PRE0 / 256), dim3(256), 0, stream>>>(out);
    // 4 b-tiles * 16 o-tiles * 8 i-segments = 512 blocks
    kan_main<<<dim3(512), dim3(256), 0, stream>>>(x, grid, coef, rs, ss, mk, out);
}